// GAT_74062416052497
// MI455X (gfx1250) — compile-verified
//
#include <hip/hip_runtime.h>
#include <stdint.h>

// ----------------------------- types ---------------------------------------
typedef float v2f __attribute__((ext_vector_type(2)));
typedef float v8f __attribute__((ext_vector_type(8)));
typedef int   v4i __attribute__((ext_vector_type(4)));

#define NEG_SLOPE 0.2f

// ------------------- optional CDNA5 async global->LDS -----------------------
#if defined(__has_builtin)
#if __has_builtin(__builtin_amdgcn_global_load_async_to_lds_b128) && \
    __has_builtin(__builtin_amdgcn_s_wait_asynccnt)
#define HAVE_ASYNC_LDS 1
#endif
#endif
#ifndef HAVE_ASYNC_LDS
#define HAVE_ASYNC_LDS 0
#endif

#if HAVE_ASYNC_LDS
typedef __attribute__((address_space(1))) v4i* g_v4i_p;  // global, 64-bit
typedef __attribute__((address_space(3))) v4i* l_v4i_p;  // LDS, 32-bit
#endif

__device__ __forceinline__ void g2l_b128(void* lds_dst, const void* g_src) {
#if HAVE_ASYNC_LDS
  __builtin_amdgcn_global_load_async_to_lds_b128(
      (g_v4i_p)(uintptr_t)g_src,
      (l_v4i_p)(uint32_t)(uintptr_t)lds_dst, 0, 0);
#else
  *reinterpret_cast<float4*>(lds_dst) = *reinterpret_cast<const float4*>(g_src);
#endif
}
__device__ __forceinline__ void g2l_wait() {
#if HAVE_ASYNC_LDS
  __builtin_amdgcn_s_wait_asynccnt(0);
#endif
}

// ----------------------------- helpers --------------------------------------
__device__ __forceinline__ float lrelu(float v) { return v > 0.f ? v : NEG_SLOPE * v; }

// float atomic max via int/uint ordering trick (init must be -inf bits)
__device__ __forceinline__ void atomicMaxF(float* a, float v) {
  if (v >= 0.f) atomicMax(reinterpret_cast<int*>(a), __float_as_int(v));
  else atomicMin(reinterpret_cast<unsigned int*>(a), (unsigned int)__float_as_int(v));
}

// ------------------------------ GEMM (WMMA) ----------------------------------
// C[n,64] = A[n,K] @ B[K,64]; fp32 via V_WMMA_F32_16X16X4_F32.
// 128 threads = 4 waves; each wave computes a 16x64 strip (4 accumulators).
// B is staged into LDS in 64x64 chunks (64 k-rows each) via async global->LDS when available.
template <int K>
__global__ __launch_bounds__(128) void gemm_n64(const float* __restrict__ A,
                                                const float* __restrict__ B,
                                                float* __restrict__ C, int n) {
  constexpr int S = 68;  // padded LDS row stride (floats), keeps 16B alignment
  __shared__ float sB[64 * S];
  const int t    = threadIdx.x;
  const int wave = t >> 5;
  const int lane = t & 31;
  const int r0   = (int)blockIdx.x * 64 + wave * 16;
  const int m16  = lane & 15;
  const int kh   = (lane >> 4) << 1;  // K sub-offset: 0 (lanes 0-15) / 2 (16-31)

  int rowA = r0 + m16;
  if (rowA > n - 1) rowA = n - 1;  // clamp: keep EXEC all-ones for WMMA
  const float* __restrict__ Arow = A + (size_t)rowA * K;

  v8f acc0 = {}, acc1 = {}, acc2 = {}, acc3 = {};

  const int rB = t >> 1;        // 0..63 : chunk row handled by this thread
  const int cB = (t & 1) * 32;  // col half

  for (int kc = 0; kc < K; kc += 64) {
    const float* gsrc = B + (size_t)(kc + rB) * 64 + cB;
    float*       ldst = &sB[rB * S + cB];
#pragma unroll
    for (int i = 0; i < 8; ++i) g2l_b128(ldst + 4 * i, gsrc + 4 * i);
    g2l_wait();
    __syncthreads();

#pragma unroll 4
    for (int kk = 0; kk < 64; kk += 4) {
      // A fragment (16x4 fp32): lane m16 holds A[m][kh], A[m][kh+1]
      v2f a;
      a.x = Arow[kc + kk + kh];
      a.y = Arow[kc + kk + kh + 1];
      const float* b0 = &sB[(kk + kh) * S + m16];
      const float* b1 = b0 + S;
      v2f b;
      b.x = b0[0];  b.y = b1[0];
      acc0 = __builtin_amdgcn_wmma_f32_16x16x4_f32(false, a, false, b, (short)0, acc0, false, false);
      b.x = b0[16]; b.y = b1[16];
      acc1 = __builtin_amdgcn_wmma_f32_16x16x4_f32(false, a, false, b, (short)0, acc1, false, false);
      b.x = b0[32]; b.y = b1[32];
      acc2 = __builtin_amdgcn_wmma_f32_16x16x4_f32(false, a, false, b, (short)0, acc2, false, false);
      b.x = b0[48]; b.y = b1[48];
      acc3 = __builtin_amdgcn_wmma_f32_16x16x4_f32(false, a, false, b, (short)0, acc3, false, false);
    }
    __syncthreads();
  }

  // D layout: VGPR j, lanes 0-15 -> M=j, lanes 16-31 -> M=8+j; N = lane%16
  const int colB = lane & 15;
  const int rOff = (lane >> 4) * 8;
#pragma unroll
  for (int j = 0; j < 8; ++j) {
    const int row = r0 + rOff + j;
    if (row < n) {
      float* Crow = C + (size_t)row * 64;
      Crow[colB + 0]  = acc0[j];
      Crow[colB + 16] = acc1[j];
      Crow[colB + 32] = acc2[j];
      Crow[colB + 48] = acc3[j];
    }
  }
}

// --------------------------- small kernels -----------------------------------
__global__ void fill_f32(float* __restrict__ p, float v, long long cnt) {
  long long i = (long long)blockIdx.x * blockDim.x + threadIdx.x;
  if (i < cnt) p[i] = v;
}

// alpha_s/alpha_d for 8 heads x 8 channels
__global__ void alpha_h8(const float* __restrict__ h, const float* __restrict__ a_s,
                         const float* __restrict__ a_d, float* __restrict__ as,
                         float* __restrict__ ad, int n) {
  int i = blockIdx.x * blockDim.x + threadIdx.x;
  if (i >= n * 8) return;
  int node = i >> 3, hd = i & 7;
  const float* hp = h + (size_t)node * 64 + hd * 8;
  float s = 0.f, d = 0.f;
#pragma unroll
  for (int c = 0; c < 8; ++c) {
    s += hp[c] * a_s[hd * 8 + c];
    d += hp[c] * a_d[hd * 8 + c];
  }
  as[i] = s;
  ad[i] = d;
}

// alpha for 1 head x 64 channels
__global__ void alpha_h1(const float* __restrict__ h, const float* __restrict__ a_s,
                         const float* __restrict__ a_d, float* __restrict__ as,
                         float* __restrict__ ad, int n) {
  int i = blockIdx.x * blockDim.x + threadIdx.x;
  if (i >= n) return;
  const float* hp = h + (size_t)i * 64;
  float s = 0.f, d = 0.f;
#pragma unroll 8
  for (int c = 0; c < 64; ++c) {
    s += hp[c] * a_s[c];
    d += hp[c] * a_d[c];
  }
  as[i] = s;
  ad[i] = d;
}

// ------------------------- edge passes, 8 heads ------------------------------
__global__ void edge_max_h8(const int* __restrict__ ei, int E, int n,
                            const float* __restrict__ as, const float* __restrict__ ad,
                            float* __restrict__ mx) {
  long long i = (long long)blockIdx.x * blockDim.x + threadIdx.x;
  long long tot = (long long)(E + n) * 8;
  if (i >= tot) return;
  int hd = (int)(i & 7);
  long long e = i >> 3;
  int src, dst;
  if (e < E) { src = ei[e]; dst = ei[(size_t)E + e]; }
  else       { src = dst = (int)(e - E); }
  float v = lrelu(as[(size_t)src * 8 + hd] + ad[(size_t)dst * 8 + hd]);
  atomicMaxF(&mx[(size_t)dst * 8 + hd], v);
}

__global__ void edge_sum_h8(const int* __restrict__ ei, int E, int n,
                            const float* __restrict__ as, const float* __restrict__ ad,
                            const float* __restrict__ mx, float* __restrict__ den) {
  long long i = (long long)blockIdx.x * blockDim.x + threadIdx.x;
  long long tot = (long long)(E + n) * 8;
  if (i >= tot) return;
  int hd = (int)(i & 7);
  long long e = i >> 3;
  int src, dst;
  if (e < E) { src = ei[e]; dst = ei[(size_t)E + e]; }
  else       { src = dst = (int)(e - E); }
  float v = lrelu(as[(size_t)src * 8 + hd] + ad[(size_t)dst * 8 + hd]);
  atomicAdd(&den[(size_t)dst * 8 + hd], __expf(v - mx[(size_t)dst * 8 + hd]));
}

__global__ void edge_agg_h8(const int* __restrict__ ei, int E, int n,
                            const float* __restrict__ as, const float* __restrict__ ad,
                            const float* __restrict__ mx, const float* __restrict__ den,
                            const float* __restrict__ h, float* __restrict__ out) {
  long long i = (long long)blockIdx.x * blockDim.x + threadIdx.x;
  long long tot = (long long)(E + n) * 8;
  if (i >= tot) return;
  int hd = (int)(i & 7);
  long long e = i >> 3;
  int src, dst;
  if (e < E) { src = ei[e]; dst = ei[(size_t)E + e]; }
  else       { src = dst = (int)(e - E); }
  const float* hs = h + (size_t)src * 64 + hd * 8;
  __builtin_prefetch(hs, 0, 0);  // global_prefetch_b8 for the gather row
  float v = lrelu(as[(size_t)src * 8 + hd] + ad[(size_t)dst * 8 + hd]);
  float w = __expf(v - mx[(size_t)dst * 8 + hd]) / (den[(size_t)dst * 8 + hd] + 1e-16f);
  float* o = out + (size_t)dst * 64 + hd * 8;
#pragma unroll
  for (int c = 0; c < 8; ++c) atomicAdd(o + c, hs[c] * w);
}

// ------------------------- edge passes, 1 head -------------------------------
__global__ void edge_max_h1(const int* __restrict__ ei, int E, int n,
                            const float* __restrict__ as, const float* __restrict__ ad,
                            float* __restrict__ mx) {
  long long i = (long long)blockIdx.x * blockDim.x + threadIdx.x;
  long long tot = (long long)E + n;
  if (i >= tot) return;
  int src, dst;
  if (i < E) { src = ei[i]; dst = ei[(size_t)E + i]; }
  else       { src = dst = (int)(i - E); }
  atomicMaxF(&mx[dst], lrelu(as[src] + ad[dst]));
}

__global__ void edge_sum_h1(const int* __restrict__ ei, int E, int n,
                            const float* __restrict__ as, const float* __restrict__ ad,
                            const float* __restrict__ mx, float* __restrict__ den) {
  long long i = (long long)blockIdx.x * blockDim.x + threadIdx.x;
  long long tot = (long long)E + n;
  if (i >= tot) return;
  int src, dst;
  if (i < E) { src = ei[i]; dst = ei[(size_t)E + i]; }
  else       { src = dst = (int)(i - E); }
  float v = lrelu(as[src] + ad[dst]);
  atomicAdd(&den[dst], __expf(v - mx[dst]));
}

__global__ void edge_agg_h1(const int* __restrict__ ei, int E, int n,
                            const float* __restrict__ as, const float* __restrict__ ad,
                            const float* __restrict__ mx, const float* __restrict__ den,
                            const float* __restrict__ h, float* __restrict__ out) {
  long long i = (long long)blockIdx.x * blockDim.x + threadIdx.x;
  long long tot = (long long)(E + n) * 4;
  if (i >= tot) return;
  int sub = (int)(i & 3);
  long long e = i >> 2;
  int src, dst;
  if (e < E) { src = ei[e]; dst = ei[(size_t)E + e]; }
  else       { src = dst = (int)(e - E); }
  const float* hs = h + (size_t)src * 64 + sub * 16;
  __builtin_prefetch(hs, 0, 0);
  float v = lrelu(as[src] + ad[dst]);
  float w = __expf(v - mx[dst]) / (den[dst] + 1e-16f);
  float* o = out + (size_t)dst * 64 + sub * 16;
#pragma unroll
  for (int c = 0; c < 16; ++c) atomicAdd(o + c, hs[c] * w);
}

// --------------------------- epilogue kernels --------------------------------
__global__ void bias_elu(float* __restrict__ x, const float* __restrict__ b,
                         long long total) {
  long long i = (long long)blockIdx.x * blockDim.x + threadIdx.x;
  if (i >= total) return;
  float v = x[i] + b[i & 63];
  x[i] = v > 0.f ? v : (__expf(v) - 1.f);
}

__global__ void pool_accum(const float* __restrict__ x, const int* __restrict__ batch,
                           float* __restrict__ pool, float* __restrict__ cnt, int n) {
  int i = blockIdx.x * blockDim.x + threadIdx.x;
  if (i >= n * 4) return;
  int node = i >> 2, sub = i & 3;
  int g = batch[node];
  const float* xs = x + (size_t)node * 64 + sub * 16;
  float* ps = pool + (size_t)g * 64 + sub * 16;
#pragma unroll
  for (int c = 0; c < 16; ++c) atomicAdd(ps + c, xs[c]);
  if (sub == 0) atomicAdd(cnt + g, 1.f);
}

__global__ void pool_final(const float* __restrict__ pool, const float* __restrict__ cnt,
                           float* __restrict__ out) {
  int i = blockIdx.x * blockDim.x + threadIdx.x;
  if (i >= 64 * 64) return;
  out[i] = pool[i] / fmaxf(cnt[i >> 6], 1.f);
}

// ------------------------------ launch ---------------------------------------
extern "C" void kernel_launch(void* const* d_in, const int* in_sizes, int n_in,
                              void* d_out, int out_size, void* d_ws, size_t ws_size,
                              hipStream_t stream) {
  const float* x    = (const float*)d_in[0];
  const int*   ei   = (const int*)d_in[1];
  const int*   batc = (const int*)d_in[2];
  const float* W1   = (const float*)d_in[3];
  const float* aS1  = (const float*)d_in[4];
  const float* aD1  = (const float*)d_in[5];
  const float* b1   = (const float*)d_in[6];
  const float* W2   = (const float*)d_in[7];
  const float* aS2  = (const float*)d_in[8];
  const float* aD2  = (const float*)d_in[9];
  const float* b2   = (const float*)d_in[10];
  float* out = (float*)d_out;

  const int n = in_sizes[0] / 512;
  const int E = in_sizes[1] / 2;

  // workspace layout (floats), with reuse of dead regions
  float* ws = (float*)d_ws;
  size_t off = 0;
  float* h1   = ws + off; off += (size_t)n * 64;  // gemm1 out; reused as agg2
  float* agg1 = ws + off; off += (size_t)n * 64;  // layer1 aggregate -> gemm2 in
  float* h2   = ws + off; off += (size_t)n * 64;  // gemm2 out
  float* aS   = ws + off; off += (size_t)n * 8;   // alpha_src (layer2 reuses [0,n))
  float* aD   = ws + off; off += (size_t)n * 8;
  float* mM   = ws + off; off += (size_t)n * 8;   // segment max
  float* dN   = ws + off; off += (size_t)n * 8;   // segment sum
  float* pool = ws + off; off += 64 * 64;
  float* cnt  = ws + off; off += 64;
  float* agg2 = h1;  // h1 dead once layer-1 aggregation is done

  const int B = 256;
  auto NB = [](long long tot, int b) { return (unsigned)((tot + b - 1) / b); };
  const float NINF = -__builtin_inff();

  // ----- layer 1 -----
  gemm_n64<512><<<(unsigned)((n + 63) / 64), 128, 0, stream>>>(x, W1, h1, n);
  alpha_h8<<<NB((long long)n * 8, B), B, 0, stream>>>(h1, aS1, aD1, aS, aD, n);

  fill_f32<<<NB((long long)n * 8, B), B, 0, stream>>>(mM, NINF, (long long)n * 8);
  (void)hipMemsetAsync(dN, 0, (size_t)n * 8 * sizeof(float), stream);
  (void)hipMemsetAsync(agg1, 0, (size_t)n * 64 * sizeof(float), stream);

  long long tot8 = (long long)(E + n) * 8;
  edge_max_h8<<<NB(tot8, B), B, 0, stream>>>(ei, E, n, aS, aD, mM);
  edge_sum_h8<<<NB(tot8, B), B, 0, stream>>>(ei, E, n, aS, aD, mM, dN);
  edge_agg_h8<<<NB(tot8, B), B, 0, stream>>>(ei, E, n, aS, aD, mM, dN, h1, agg1);
  bias_elu<<<NB((long long)n * 64, B), B, 0, stream>>>(agg1, b1, (long long)n * 64);

  // ----- layer 2 -----
  gemm_n64<64><<<(unsigned)((n + 63) / 64), 128, 0, stream>>>(agg1, W2, h2, n);
  alpha_h1<<<NB(n, B), B, 0, stream>>>(h2, aS2, aD2, aS, aD, n);

  fill_f32<<<NB(n, B), B, 0, stream>>>(mM, NINF, (long long)n);
  (void)hipMemsetAsync(dN, 0, (size_t)n * sizeof(float), stream);
  (void)hipMemsetAsync(agg2, 0, (size_t)n * 64 * sizeof(float), stream);

  long long tot1 = (long long)E + n;
  edge_max_h1<<<NB(tot1, B), B, 0, stream>>>(ei, E, n, aS, aD, mM);
  edge_sum_h1<<<NB(tot1, B), B, 0, stream>>>(ei, E, n, aS, aD, mM, dN);
  edge_agg_h1<<<NB(tot1 * 4, B), B, 0, stream>>>(ei, E, n, aS, aD, mM, dN, h2, agg2);
  bias_elu<<<NB((long long)n * 64, B), B, 0, stream>>>(agg2, b2, (long long)n * 64);

  // ----- global mean pool -----
  (void)hipMemsetAsync(pool, 0, (64 * 64 + 64) * sizeof(float), stream);  // pool + cnt
  pool_accum<<<NB((long long)n * 4, B), B, 0, stream>>>(agg2, batc, pool, cnt, n);
  pool_final<<<NB(64 * 64, B), B, 0, stream>>>(pool, cnt, out);
}